// GCN_18459769438249
// MI455X (gfx1250) — compile-verified
//
#include <hip/hip_runtime.h>
#include <hip/hip_bf16.h>

#define NFEAT 256
#define NHID1 128
#define NHID2 64
#define NCLASS 200

typedef __attribute__((ext_vector_type(2))) float v2f;
typedef __attribute__((ext_vector_type(8))) float v8f;

// ---------------------------------------------------------------------------
// Degree accumulation: deg_src[src[e]] += 1, deg_dst[dst[e]] += 1
// ---------------------------------------------------------------------------
__global__ __launch_bounds__(256) void degree_kernel(
    const int* __restrict__ src, const int* __restrict__ dst,
    float* __restrict__ deg_src, float* __restrict__ deg_dst, int E) {
  int e = blockIdx.x * blockDim.x + threadIdx.x;
  if (e < E) {
    atomicAdd(&deg_src[src[e]], 1.0f);
    atomicAdd(&deg_dst[dst[e]], 1.0f);
  }
}

// deg -> rsqrt(max(deg, 1))   (in place, covers both deg arrays contiguously)
__global__ __launch_bounds__(256) void dinv_kernel(float* __restrict__ deg, int n) {
  int i = blockIdx.x * blockDim.x + threadIdx.x;
  if (i < n) deg[i] = rsqrtf(fmaxf(deg[i], 1.0f));
}

// ---------------------------------------------------------------------------
// WMMA fp32 GEMM:  C[M,NCOLS] = A[M,K] @ B[K,NCOLS] (optionally *rowscale, +bias)
// One wave computes a 16-row strip across all NTILES 16-column tiles using
// V_WMMA_F32_16X16X4_F32 (full fp32 precision, matches the reference math).
// ---------------------------------------------------------------------------
template <int K, int NTILES, int NCOLS, bool SCALE_ROWS, bool ADD_BIAS>
__global__ __launch_bounds__(256) void gemm_wmma_kernel(
    const float* __restrict__ A, const float* __restrict__ B,
    const float* __restrict__ bias, const float* __restrict__ rowscale,
    float* __restrict__ C, int M) {
  static_assert(NTILES * 16 >= NCOLS, "tiles must cover NCOLS");
  const int wave = blockIdx.x * (blockDim.x >> 5) + (threadIdx.x >> 5);
  const int row0 = wave * 16;
  if (row0 >= M) return;                 // wave-uniform exit
  const int lane = threadIdx.x & 31;
  const int half = lane >> 4;            // selects K pair {0,1} vs {2,3}
  const int r    = lane & 15;            // M index (A) / N index (B,C)

  if (row0 + 16 <= M) {
    v8f acc[NTILES] = {};
    const float* arow = A + (size_t)(row0 + r) * K;
#pragma unroll 2
    for (int k0 = 0; k0 < K; k0 += 4) {
      const int ka = k0 + 2 * half;
      v2f a;
      a.x = arow[ka];                    // contiguous 8B per lane
      a.y = arow[ka + 1];
      const float* b0 = B + (size_t)ka * NCOLS;
#pragma unroll
      for (int t = 0; t < NTILES; ++t) {
        const int col = t * 16 + r;
        v2f b;
        if ((NCOLS % 16 == 0) || (col < NCOLS)) {
          b.x = b0[col];
          b.y = b0[NCOLS + col];
        } else {
          b.x = 0.0f;
          b.y = 0.0f;
        }
        acc[t] = __builtin_amdgcn_wmma_f32_16x16x4_f32(
            false, a, false, b, (short)0, acc[t], false, false);
      }
    }

    float rs[8];
#pragma unroll
    for (int v = 0; v < 8; ++v)
      rs[v] = SCALE_ROWS ? rowscale[row0 + v + 8 * half] : 1.0f;

#pragma unroll
    for (int t = 0; t < NTILES; ++t) {
      const int col = t * 16 + r;
      if ((NCOLS % 16 == 0) || (col < NCOLS)) {
        const float bv = ADD_BIAS ? bias[col] : 0.0f;
#pragma unroll
        for (int v = 0; v < 8; ++v) {
          const int m = row0 + v + 8 * half;
          C[(size_t)m * NCOLS + col] = acc[t][v] * rs[v] + bv;
        }
      }
    }
  } else {
    // scalar tail for a partial M tile (not hit when M % 16 == 0)
    for (int m = row0 + lane; m < M; m += 32) {
      const float rsv = SCALE_ROWS ? rowscale[m] : 1.0f;
      for (int col = 0; col < NCOLS; ++col) {
        float s = 0.0f;
        for (int k = 0; k < K; ++k)
          s += A[(size_t)m * K + k] * B[(size_t)k * NCOLS + col];
        C[(size_t)m * NCOLS + col] = s * rsv + (ADD_BIAS ? bias[col] : 0.0f);
      }
    }
  }
}

// ---------------------------------------------------------------------------
// SpMM aggregate: out[dst[e], :] += H[src[e], :]   (edge-parallel, float4)
// ---------------------------------------------------------------------------
template <int F>
__global__ __launch_bounds__(256) void spmm_kernel(
    const int* __restrict__ src, const int* __restrict__ dst,
    const float* __restrict__ H, float* __restrict__ out, int E) {
  constexpr int TPE = F / 4;        // threads per edge, each handles float4
  constexpr int EPB = 256 / TPE;    // edges per block
  const int e = blockIdx.x * EPB + (int)(threadIdx.x / TPE);
  if (e >= E) return;
  const int f4 = (threadIdx.x % TPE) * 4;
  const int s = src[e];
  const int d = dst[e];
  const float4 v = *reinterpret_cast<const float4*>(&H[(size_t)s * F + f4]);
  float* o = &out[(size_t)d * F + f4];
  atomicAdd(o + 0, v.x);
  atomicAdd(o + 1, v.y);
  atomicAdd(o + 2, v.z);
  atomicAdd(o + 3, v.w);
}

// ---------------------------------------------------------------------------
// Post-aggregation: h = relu(agg * dinv_dst[row] + bias[col])   (in place)
// ---------------------------------------------------------------------------
template <int F>
__global__ __launch_bounds__(256) void post_kernel(
    float* __restrict__ agg, const float* __restrict__ dinv_dst,
    const float* __restrict__ bias, int M) {
  const long long i = (long long)blockIdx.x * blockDim.x + threadIdx.x;
  if (i < (long long)M * F) {
    const int row = (int)(i / F);
    const int col = (int)(i % F);
    agg[i] = fmaxf(agg[i] * dinv_dst[row] + bias[col], 0.0f);
  }
}

// ---------------------------------------------------------------------------
// Row-wise log-softmax (one wave32 per row, shuffle reductions), in place.
// ---------------------------------------------------------------------------
__global__ __launch_bounds__(256) void log_softmax_kernel(
    float* __restrict__ out, int M, int C) {
  const int row = blockIdx.x * (blockDim.x >> 5) + (threadIdx.x >> 5);
  if (row >= M) return;
  const int lane = threadIdx.x & 31;
  float* p = out + (size_t)row * C;

  float mx = -3.402823466e+38f;
  for (int c = lane; c < C; c += 32) mx = fmaxf(mx, p[c]);
#pragma unroll
  for (int off = 16; off > 0; off >>= 1) mx = fmaxf(mx, __shfl_xor(mx, off, 32));

  float s = 0.0f;
  for (int c = lane; c < C; c += 32) s += expf(p[c] - mx);
#pragma unroll
  for (int off = 16; off > 0; off >>= 1) s += __shfl_xor(s, off, 32);

  const float lse = mx + logf(s);
  for (int c = lane; c < C; c += 32) p[c] -= lse;
}

// ---------------------------------------------------------------------------
// Launcher
// ---------------------------------------------------------------------------
extern "C" void kernel_launch(void* const* d_in, const int* in_sizes, int n_in,
                              void* d_out, int out_size, void* d_ws, size_t ws_size,
                              hipStream_t stream) {
  const float* x   = (const float*)d_in[0];
  const int*   src = (const int*)d_in[1];
  const int*   dst = (const int*)d_in[2];
  const float* W1  = (const float*)d_in[3];
  const float* b1  = (const float*)d_in[4];
  const float* W2  = (const float*)d_in[5];
  const float* b2  = (const float*)d_in[6];
  const float* Wf  = (const float*)d_in[7];
  const float* bf  = (const float*)d_in[8];
  float* out = (float*)d_out;
  (void)n_in; (void)out_size; (void)ws_size;

  const int M = in_sizes[0] / NFEAT;   // 100000
  const int E = in_sizes[1];           // 1700000

  auto alignup = [](size_t v) { return (v + 255) & ~(size_t)255; };
  char* ws = (char*)d_ws;
  size_t o = 0;
  float* deg = (float*)(ws + o); o = alignup(o + (size_t)2 * M * sizeof(float));
  float* t1  = (float*)(ws + o); o = alignup(o + (size_t)M * NHID1 * sizeof(float));
  float* h1  = (float*)(ws + o); o = alignup(o + (size_t)M * NHID1 * sizeof(float));
  float* t2  = (float*)(ws + o); o = alignup(o + (size_t)M * NHID2 * sizeof(float));
  float* agg2 = t1;                 // t1 dead after SpMM1 -> reuse for layer-2 agg
  float* dinv_src = deg;
  float* dinv_dst = deg + M;

  // --- degrees -> D^{-1/2} ---
  hipMemsetAsync(deg, 0, (size_t)2 * M * sizeof(float), stream);
  degree_kernel<<<(E + 255) / 256, 256, 0, stream>>>(src, dst, dinv_src, dinv_dst, E);
  dinv_kernel<<<(2 * M + 255) / 256, 256, 0, stream>>>(deg, 2 * M);

  const int waves = (M + 15) / 16;
  const int gblk  = (waves + 7) / 8;   // 8 waves (256 threads) per block

  // --- layer 1: t1 = (x @ W1) * dinv_src ; agg ; h1 = relu(agg*dinv_dst + b1)
  gemm_wmma_kernel<NFEAT, NHID1 / 16, NHID1, true, false>
      <<<gblk, 256, 0, stream>>>(x, W1, nullptr, dinv_src, t1, M);
  hipMemsetAsync(h1, 0, (size_t)M * NHID1 * sizeof(float), stream);
  {
    constexpr int EPB = 256 / (NHID1 / 4);
    spmm_kernel<NHID1><<<(E + EPB - 1) / EPB, 256, 0, stream>>>(src, dst, t1, h1, E);
  }
  post_kernel<NHID1><<<(unsigned)(((size_t)M * NHID1 + 255) / 256), 256, 0, stream>>>(
      h1, dinv_dst, b1, M);

  // --- layer 2 ---
  gemm_wmma_kernel<NHID1, NHID2 / 16, NHID2, true, false>
      <<<gblk, 256, 0, stream>>>(h1, W2, nullptr, dinv_src, t2, M);
  hipMemsetAsync(agg2, 0, (size_t)M * NHID2 * sizeof(float), stream);
  {
    constexpr int EPB = 256 / (NHID2 / 4);
    spmm_kernel<NHID2><<<(E + EPB - 1) / EPB, 256, 0, stream>>>(src, dst, t2, agg2, E);
  }
  post_kernel<NHID2><<<(unsigned)(((size_t)M * NHID2 + 255) / 256), 256, 0, stream>>>(
      agg2, dinv_dst, b2, M);

  // --- classifier: logits -> d_out, then in-place log-softmax ---
  gemm_wmma_kernel<NHID2, 13, NCLASS, false, true>
      <<<gblk, 256, 0, stream>>>(agg2, Wf, bf, nullptr, out, M);
  log_softmax_kernel<<<(M + 7) / 8, 256, 0, stream>>>(out, M, NCLASS);
}